// TransformerEncoder_84722524881278
// MI455X (gfx1250) — compile-verified
//
#include <hip/hip_runtime.h>
#include <math.h>

// Sizes from reference
#define BB 8
#define NP 2048
#define KN 16
#define NL 4
#define NH 8
#define PD 256
#define FFD 512
#define WD 512
#define SQ 8

typedef __attribute__((ext_vector_type(2))) float v2f;
typedef __attribute__((ext_vector_type(8))) float v8f;

__device__ __forceinline__ float lrelu(float v) { return v > 0.f ? v : 0.2f * v; }

// ---------------------------------------------------------------------------
// KNN: per query point, keep top-16 of d = 2*q.m - |q|^2 - |m|^2 (largest)
// Fully static-indexed insertion (registers, no scratch). Stable ties -> matches top_k.
// ---------------------------------------------------------------------------
__global__ void knn_kernel(const float* __restrict__ x, int* __restrict__ idx)
{
  int n = blockIdx.x * blockDim.x + threadIdx.x;
  int b = blockIdx.y;
  const float* xb = x + (size_t)b * NP * 3;
  float qx = xb[3*n+0], qy = xb[3*n+1], qz = xb[3*n+2];
  float qq = qx*qx + qy*qy + qz*qz;
  float bv[KN]; int bi[KN];
#pragma unroll
  for (int i = 0; i < KN; ++i) { bv[i] = -INFINITY; bi[i] = 0; }
  for (int m = 0; m < NP; ++m) {
    float mx = xb[3*m], my = xb[3*m+1], mz = xb[3*m+2];
    float d = 2.f*(qx*mx + qy*my + qz*mz) - qq - (mx*mx + my*my + mz*mz);
    if (d > bv[KN-1]) {
      bv[KN-1] = d; bi[KN-1] = m;
#pragma unroll
      for (int i = KN-1; i > 0; --i) {
        if (bv[i] > bv[i-1]) {
          float tv = bv[i]; bv[i] = bv[i-1]; bv[i-1] = tv;
          int   ti = bi[i]; bi[i] = bi[i-1]; bi[i-1] = ti;
        }
      }
    }
  }
  int* op = idx + ((size_t)b * NP + n) * KN;
#pragma unroll
  for (int i = 0; i < KN; ++i) op[i] = bi[i];
}

// ---------------------------------------------------------------------------
// EdgeConv layer 0: h0[b,o,n] = max_k lrelu(W[:,0:3].(nb-y) + W[:,3:6].y + b)
// block = 64 threads (one per output channel), grid (N, B). Writes hcat rows 0..63.
// ---------------------------------------------------------------------------
__global__ void edgeconv_kernel(const float* __restrict__ x, const int* __restrict__ idx,
                                const float* __restrict__ W, const float* __restrict__ bias,
                                float* __restrict__ hcat)
{
  int o = threadIdx.x, n = blockIdx.x, b = blockIdx.y;
  const float* xb = x + (size_t)b * NP * 3;
  float yx = xb[3*n], yy = xb[3*n+1], yz = xb[3*n+2];
  float w0 = W[o*6+0], w1 = W[o*6+1], w2 = W[o*6+2];
  float w3 = W[o*6+3], w4 = W[o*6+4], w5 = W[o*6+5];
  float base = w3*yx + w4*yy + w5*yz + bias[o];
  const int* ip = idx + ((size_t)b * NP + n) * KN;
  float best = -INFINITY;
#pragma unroll
  for (int k = 0; k < KN; ++k) {
    int j = ip[k];
    float v = w0*(xb[3*j]-yx) + w1*(xb[3*j+1]-yy) + w2*(xb[3*j+2]-yz) + base;
    best = fmaxf(best, lrelu(v));
  }
  hcat[((size_t)b * WD + o) * NP + n] = best;
}

// ---------------------------------------------------------------------------
// Graph max pooling: pooled[b,c,n] = max_k hcat[b,c_off+c, idx[b,n,k]]
// block = C threads, grid (N, B). pooled layout [B,C,NP].
// ---------------------------------------------------------------------------
__global__ void gather_max_kernel(const float* __restrict__ hcat, const int* __restrict__ idx,
                                  float* __restrict__ pooled, int c_off)
{
  int c = threadIdx.x, n = blockIdx.x, b = blockIdx.y;
  int C = blockDim.x;
  const int* ip = idx + ((size_t)b * NP + n) * KN;
  const float* row = hcat + ((size_t)b * WD + c_off + c) * NP;
  float best = -INFINITY;
#pragma unroll
  for (int k = 0; k < KN; ++k) best = fmaxf(best, row[ip[k]]);
  pooled[((size_t)b * C + c) * NP + n] = best;
}

// ---------------------------------------------------------------------------
// WMMA fp32 GEMM:  C[M,Ncols] = A[M,K](rm) * B[K,Ncols](rm)  (+bias, +leakyrelu)
// Block = 4 waves (128 thr). Each wave owns a 16x64 C tile (4 v8f accumulators),
// so one A fragment feeds 4 V_WMMA_F32_16X16X4_F32 per K-step; all 4 waves in the
// block share the same A rows (L0/L2 reuse). Block tile = 16 x 256.
// Fragment layouts per ISA 7.12.2:
//   A 16x4 : lanes 0-15 hold row=lane, K={k,k+1}; lanes 16-31 row=lane-16, K={k+2,k+3}
//   B 4x16 : lanes 0-15 hold col=lane, rows {k,k+1}; lanes 16-31 col=lane-16, rows {k+2,k+3}
//   C 16x16: VGPR i -> row (lane<16 ? i : 8+i), col = lane&15
// EXEC all ones (exact grids), M%16==0, Ncols%256==0, K%4==0.
// bias_mode: 0 none, 1 per output row, 2 per output col.  batch via blockIdx.z strides.
// ---------------------------------------------------------------------------
__global__ void gemm_f32_wmma(const float* __restrict__ A, const float* __restrict__ Bm,
                              float* __restrict__ C, const float* __restrict__ bias,
                              int M, int Ncols, int Kd,
                              long sA, long sB, long sC, int bias_mode, int do_act)
{
  int lane = threadIdx.x & 31;
  int wave = threadIdx.x >> 5;
  int half = lane >> 4;
  int l16  = lane & 15;
  int nb = blockIdx.x * 256 + wave * 64;
  int mb = blockIdx.y * 16;
  const float* Ab = A  + (size_t)blockIdx.z * sA;
  const float* Bb = Bm + (size_t)blockIdx.z * sB;
  float*       Cb = C  + (size_t)blockIdx.z * sC;

  v8f acc0 = {0.f,0.f,0.f,0.f,0.f,0.f,0.f,0.f};
  v8f acc1 = acc0, acc2 = acc0, acc3 = acc0;
  const float* ap = Ab + (size_t)(mb + l16) * Kd + half * 2;
  const float* bp = Bb + (size_t)(half * 2) * Ncols + nb + l16;
  for (int k = 0; k < Kd; k += 4) {
    __builtin_prefetch(ap + 32, 0, 1);          // global_prefetch_b8: next A K-chunk
    v2f a;  a.x  = ap[0];      a.y  = ap[1];
    v2f b0; b0.x = bp[0];      b0.y = bp[Ncols];
    v2f b1; b1.x = bp[16];     b1.y = bp[Ncols + 16];
    v2f b2; b2.x = bp[32];     b2.y = bp[Ncols + 32];
    v2f b3; b3.x = bp[48];     b3.y = bp[Ncols + 48];
    acc0 = __builtin_amdgcn_wmma_f32_16x16x4_f32(false, a, false, b0, (short)0, acc0, false, false);
    acc1 = __builtin_amdgcn_wmma_f32_16x16x4_f32(false, a, false, b1, (short)0, acc1, false, false);
    acc2 = __builtin_amdgcn_wmma_f32_16x16x4_f32(false, a, false, b2, (short)0, acc2, false, false);
    acc3 = __builtin_amdgcn_wmma_f32_16x16x4_f32(false, a, false, b3, (short)0, acc3, false, false);
    ap += 4;
    bp += (size_t)4 * Ncols;
  }
#pragma unroll
  for (int i = 0; i < 8; ++i) {
    int row = mb + half * 8 + i;
    float rb = (bias_mode == 1) ? bias[row] : 0.f;
    float* crow = Cb + (size_t)row * Ncols + nb + l16;
    float v0 = acc0[i] + rb, v1 = acc1[i] + rb, v2 = acc2[i] + rb, v3 = acc3[i] + rb;
    if (bias_mode == 2) {
      v0 += bias[nb + l16];      v1 += bias[nb + 16 + l16];
      v2 += bias[nb + 32 + l16]; v3 += bias[nb + 48 + l16];
    }
    if (do_act) { v0 = lrelu(v0); v1 = lrelu(v1); v2 = lrelu(v2); v3 = lrelu(v3); }
    crow[0] = v0; crow[16] = v1; crow[32] = v2; crow[48] = v3;
  }
}

// 256x256 weight transpose (for out = A @ W^T via row-major GEMM)
__global__ void transpose256_kernel(const float* __restrict__ W, float* __restrict__ WT)
{
  int t = blockIdx.x * 256 + threadIdx.x;
  int r = t >> 8, c = t & 255;
  WT[(size_t)c * 256 + r] = W[t];
}

// max over points: xm[b,c] = max_n hf[b,c,n].  one wave per (c,b).
__global__ void rowmax_kernel(const float* __restrict__ hf, float* __restrict__ xm)
{
  int c = blockIdx.x, b = blockIdx.y, lane = threadIdx.x;
  const float* row = hf + ((size_t)b * WD + c) * NP;
  float m = -INFINITY;
  for (int n = lane; n < NP; n += 32) m = fmaxf(m, row[n]);
#pragma unroll
  for (int off = 16; off; off >>= 1) m = fmaxf(m, __shfl_xor(m, off));
  if (lane == 0) xm[(size_t)b * WD + c] = m;
}

// Generic small projection: out[bs,o] = act(sum_k in[bs,k]*W[o,k] + bias[o])
__global__ void rowproj_kernel(const float* __restrict__ in, const float* __restrict__ W,
                               const float* __restrict__ bias, float* __restrict__ out,
                               int Kd, int O, int act, int total)
{
  int t = blockIdx.x * blockDim.x + threadIdx.x;
  if (t >= total) return;
  int o = t % O, bs = t / O;
  const float* ip = in + (size_t)bs * Kd;
  const float* wr = W + (size_t)o * Kd;
  float v = bias[o];
  for (int k = 0; k < Kd; ++k) v += ip[k] * wr[k];
  if (act) v = lrelu(v);
  out[(size_t)bs * O + o] = v;
}

// Self-attention over the 8 query codes: one wave per (qi,h,b), lane = head dim.
__global__ void attn_small_kernel(const float* __restrict__ qh, const float* __restrict__ kh,
                                  const float* __restrict__ vh, float* __restrict__ out)
{
  int qi = blockIdx.x, h = blockIdx.y, b = blockIdx.z, d = threadIdx.x;
  float qd = qh[((size_t)(b*SQ) + qi) * PD + h*32 + d];
  float s[SQ];
#pragma unroll
  for (int j = 0; j < SQ; ++j) {
    float p = qd * kh[((size_t)(b*SQ) + j) * PD + h*32 + d];
#pragma unroll
    for (int off = 16; off; off >>= 1) p += __shfl_xor(p, off);
    s[j] = p * 0.17677669529663687f;   // 1/sqrt(32)
  }
  float m = s[0];
#pragma unroll
  for (int j = 1; j < SQ; ++j) m = fmaxf(m, s[j]);
  float l = 0.f;
#pragma unroll
  for (int j = 0; j < SQ; ++j) { s[j] = __expf(s[j] - m); l += s[j]; }
  float od = 0.f;
#pragma unroll
  for (int j = 0; j < SQ; ++j) od += s[j] * vh[((size_t)(b*SQ) + j) * PD + h*32 + d];
  out[((size_t)(b*SQ) + qi) * PD + h*32 + d] = od / l;
}

// Cross-attention: 8 queries x 2048 keys. One wave per (qi,h,b); scores in LDS.
__global__ void attn_cross_kernel(const float* __restrict__ qh, const float* __restrict__ kmem,
                                  const float* __restrict__ vmem, float* __restrict__ out)
{
  __shared__ float sc[NP];
  __shared__ float lq[32];
  int qi = blockIdx.x, h = blockIdx.y, b = blockIdx.z, lane = threadIdx.x;
  lq[lane] = qh[((size_t)(b*SQ) + qi) * PD + h*32 + lane];
  __syncthreads();
  const float* kb = kmem + (size_t)b * NP * PD + h*32;
  float mloc = -INFINITY;
  for (int j = lane; j < NP; j += 32) {
    const float* kr = kb + (size_t)j * PD;
    float s = 0.f;
#pragma unroll
    for (int d = 0; d < 32; ++d) s += lq[d] * kr[d];
    s *= 0.17677669529663687f;
    sc[j] = s;
    mloc = fmaxf(mloc, s);
  }
#pragma unroll
  for (int off = 16; off; off >>= 1) mloc = fmaxf(mloc, __shfl_xor(mloc, off));
  float lloc = 0.f;
  for (int j = lane; j < NP; j += 32) { float e = __expf(sc[j] - mloc); sc[j] = e; lloc += e; }
#pragma unroll
  for (int off = 16; off; off >>= 1) lloc += __shfl_xor(lloc, off);
  __syncthreads();
  const float* vb = vmem + (size_t)b * NP * PD + h*32 + lane;
  float od = 0.f;
  for (int j = 0; j < NP; ++j) od += sc[j] * vb[(size_t)j * PD];
  out[((size_t)(b*SQ) + qi) * PD + h*32 + lane] = od / lloc;
}

// Fused: out = LN(res + in @ W^T + bias). block = 256 threads per (s,b). in-place on res ok.
__global__ void proj_res_ln_kernel(const float* __restrict__ in, const float* __restrict__ W,
                                   const float* __restrict__ bias, const float* __restrict__ res,
                                   const float* __restrict__ g, const float* __restrict__ be,
                                   float* __restrict__ out, int Kd)
{
  __shared__ float red[PD];
  int s = blockIdx.x, b = blockIdx.y, o = threadIdx.x;
  size_t bs = (size_t)(b*SQ) + s;
  const float* ip = in + bs * Kd;
  const float* wr = W + (size_t)o * Kd;
  float v = bias[o];
  for (int k = 0; k < Kd; ++k) v += ip[k] * wr[k];
  v += res[bs * PD + o];
  red[o] = v; __syncthreads();
  for (int st = 128; st; st >>= 1) { if (o < st) red[o] += red[o + st]; __syncthreads(); }
  float mean = red[0] * (1.f / PD); __syncthreads();
  float dv = v - mean;
  red[o] = dv * dv; __syncthreads();
  for (int st = 128; st; st >>= 1) { if (o < st) red[o] += red[o + st]; __syncthreads(); }
  float var = red[0] * (1.f / PD);
  out[bs * PD + o] = g[o] * dv * rsqrtf(var + 1e-5f) + be[o];
}

// Final LN + Wcm projection -> d_out[b, s*64+o]
__global__ void final_kernel(const float* __restrict__ q, const float* __restrict__ g,
                             const float* __restrict__ be, const float* __restrict__ Wcm,
                             const float* __restrict__ bcm, float* __restrict__ out)
{
  __shared__ float red[PD];
  __shared__ float qn[PD];
  int s = blockIdx.x, b = blockIdx.y, o = threadIdx.x;
  size_t bs = (size_t)(b*SQ) + s;
  float v = q[bs * PD + o];
  red[o] = v; __syncthreads();
  for (int st = 128; st; st >>= 1) { if (o < st) red[o] += red[o + st]; __syncthreads(); }
  float mean = red[0] * (1.f / PD); __syncthreads();
  float dv = v - mean;
  red[o] = dv * dv; __syncthreads();
  for (int st = 128; st; st >>= 1) { if (o < st) red[o] += red[o + st]; __syncthreads(); }
  float var = red[0] * (1.f / PD);
  qn[o] = g[o] * dv * rsqrtf(var + 1e-5f) + be[o];
  __syncthreads();
  if (o < 64) {
    float acc = bcm[o];
    const float* wr = Wcm + (size_t)o * PD;
    for (int k = 0; k < PD; ++k) acc += qn[k] * wr[k];
    out[(size_t)b * WD + s * 64 + o] = acc;
  }
}

// ---------------------------------------------------------------------------
extern "C" void kernel_launch(void* const* d_in, const int* in_sizes, int n_in,
                              void* d_out, int out_size, void* d_ws, size_t ws_size,
                              hipStream_t stream)
{
  (void)in_sizes; (void)n_in; (void)out_size; (void)ws_size;
  const float* x    = (const float*)d_in[0];
  const float* W_ec = (const float*)d_in[1];
  const float* b_ec = (const float*)d_in[2];
  const float* Wc1  = (const float*)d_in[3];
  const float* bc1  = (const float*)d_in[4];
  const float* Wc2  = (const float*)d_in[5];
  const float* bc2  = (const float*)d_in[6];
  const float* Wc3  = (const float*)d_in[7];
  const float* bc3  = (const float*)d_in[8];
  const float* Wf   = (const float*)d_in[9];
  const float* bf   = (const float*)d_in[10];
  const float* Wpc  = (const float*)d_in[11];
  const float* bpc  = (const float*)d_in[12];
  const float* Wpi  = (const float*)d_in[13];
  const float* bpi  = (const float*)d_in[14];
  const float* Wq   = (const float*)d_in[15];
  const float* bq   = (const float*)d_in[16];
  const float* Wk   = (const float*)d_in[17];
  const float* bk   = (const float*)d_in[18];
  const float* Wv   = (const float*)d_in[19];
  const float* bv   = (const float*)d_in[20];
  const float* Wo   = (const float*)d_in[21];
  const float* bo   = (const float*)d_in[22];
  const float* ln1g = (const float*)d_in[23];
  const float* ln1b = (const float*)d_in[24];
  const float* ln2g = (const float*)d_in[25];
  const float* ln2b = (const float*)d_in[26];
  const float* Wff1 = (const float*)d_in[27];
  const float* bff1 = (const float*)d_in[28];
  const float* Wff2 = (const float*)d_in[29];
  const float* bff2 = (const float*)d_in[30];
  const float* ln3g = (const float*)d_in[31];
  const float* ln3b = (const float*)d_in[32];
  const float* lnfg = (const float*)d_in[33];
  const float* lnfb = (const float*)d_in[34];
  const float* Wcm  = (const float*)d_in[35];
  const float* bcm  = (const float*)d_in[36];

  // Workspace layout (floats). Peak ~74 MB. mem/kmem/vmem reuse hcat/hf (dead after xm).
  float* ws     = (float*)d_ws;
  int*   idx    = (int*)d_ws;             // B*N*K               =   262,144
  float* hcat   = ws + 262144;            // [B,512,N]           = 8,388,608
  float* pooled = hcat + 8388608;         // [B,<=128,N]         = 2,097,152
  float* hf     = pooled + 2097152;       // [B,512,N]           = 8,388,608
  float* xm     = hf + 8388608;           // [B,512]
  float* WkT    = xm + 4096;              // 256*256
  float* WvT    = WkT + 65536;
  float* q      = WvT + 65536;            // [B,8,256]
  float* qh     = q + 16384;
  float* kh     = qh + 16384;
  float* vh     = kh + 16384;
  float* attn   = vh + 16384;
  float* ffbuf  = attn + 16384;           // [B,8,512]
  float* memb   = hcat;                   // [B,N,256] reuse
  float* kmem   = hcat + 4194304;         // [B,N,256] reuse
  float* vmem   = hf;                     // [B,N,256] reuse

  // --- point-cloud encoder ---
  knn_kernel<<<dim3(NP/128, BB), 128, 0, stream>>>(x, idx);
  edgeconv_kernel<<<dim3(NP, BB), 64, 0, stream>>>(x, idx, W_ec, b_ec, hcat);

  // conv1: in ch 0..63 -> ch 64..127
  gather_max_kernel<<<dim3(NP, BB), 64, 0, stream>>>(hcat, idx, pooled, 0);
  gemm_f32_wmma<<<dim3(NP/256, 64/16, BB), 128, 0, stream>>>(
      Wc1, pooled, hcat + (size_t)64*NP, bc1, 64, NP, 64,
      0L, (long)64*NP, (long)WD*NP, 1, 1);
  // conv2: in ch 64..127 -> ch 128..255
  gather_max_kernel<<<dim3(NP, BB), 64, 0, stream>>>(hcat, idx, pooled, 64);
  gemm_f32_wmma<<<dim3(NP/256, 128/16, BB), 128, 0, stream>>>(
      Wc2, pooled, hcat + (size_t)128*NP, bc2, 128, NP, 64,
      0L, (long)64*NP, (long)WD*NP, 1, 1);
  // conv3: in ch 128..255 -> ch 256..511
  gather_max_kernel<<<dim3(NP, BB), 128, 0, stream>>>(hcat, idx, pooled, 128);
  gemm_f32_wmma<<<dim3(NP/256, 256/16, BB), 128, 0, stream>>>(
      Wc3, pooled, hcat + (size_t)256*NP, bc3, 256, NP, 128,
      0L, (long)128*NP, (long)WD*NP, 1, 1);
  // Wf: [512x512] @ hcat -> hf (bias, no act)
  gemm_f32_wmma<<<dim3(NP/256, WD/16, BB), 128, 0, stream>>>(
      Wf, hcat, hf, bf, WD, NP, WD,
      0L, (long)WD*NP, (long)WD*NP, 1, 0);
  rowmax_kernel<<<dim3(WD, BB), 32, 0, stream>>>(hf, xm);

  // q0 = lrelu(xm.reshape(B,8,64) @ Wpc^T + bpc); mem = lrelu(x @ Wpi^T + bpi)
  rowproj_kernel<<<(BB*SQ*PD)/256, 256, 0, stream>>>(xm, Wpc, bpc, q, 64, PD, 1, BB*SQ*PD);
  rowproj_kernel<<<(BB*NP*PD)/256, 256, 0, stream>>>(x, Wpi, bpi, memb, 3, PD, 1, BB*NP*PD);

  // --- perceiver decoder ---
  for (int l = 0; l < NL; ++l) {
    const float* Wq0 = Wq + (size_t)(l*2+0)*PD*PD; const float* bq0 = bq + (l*2+0)*PD;
    const float* Wk0 = Wk + (size_t)(l*2+0)*PD*PD; const float* bk0 = bk + (l*2+0)*PD;
    const float* Wv0 = Wv + (size_t)(l*2+0)*PD*PD; const float* bv0 = bv + (l*2+0)*PD;
    const float* Wo0 = Wo + (size_t)(l*2+0)*PD*PD; const float* bo0 = bo + (l*2+0)*PD;
    const float* Wq1 = Wq + (size_t)(l*2+1)*PD*PD; const float* bq1 = bq + (l*2+1)*PD;
    const float* Wk1 = Wk + (size_t)(l*2+1)*PD*PD; const float* bk1 = bk + (l*2+1)*PD;
    const float* Wv1 = Wv + (size_t)(l*2+1)*PD*PD; const float* bv1 = bv + (l*2+1)*PD;
    const float* Wo1 = Wo + (size_t)(l*2+1)*PD*PD; const float* bo1 = bo + (l*2+1)*PD;

    // self-attention over q (tiny)
    rowproj_kernel<<<(BB*SQ*PD)/256, 256, 0, stream>>>(q, Wq0, bq0, qh, PD, PD, 0, BB*SQ*PD);
    rowproj_kernel<<<(BB*SQ*PD)/256, 256, 0, stream>>>(q, Wk0, bk0, kh, PD, PD, 0, BB*SQ*PD);
    rowproj_kernel<<<(BB*SQ*PD)/256, 256, 0, stream>>>(q, Wv0, bv0, vh, PD, PD, 0, BB*SQ*PD);
    attn_small_kernel<<<dim3(SQ, NH, BB), 32, 0, stream>>>(qh, kh, vh, attn);
    proj_res_ln_kernel<<<dim3(SQ, BB), PD, 0, stream>>>(attn, Wo0, bo0, q,
                                                        ln1g + l*PD, ln1b + l*PD, q, PD);
    // cross-attention: big K/V projections via WMMA (out = mem @ W^T)
    transpose256_kernel<<<256, 256, 0, stream>>>(Wk1, WkT);
    transpose256_kernel<<<256, 256, 0, stream>>>(Wv1, WvT);
    rowproj_kernel<<<(BB*SQ*PD)/256, 256, 0, stream>>>(q, Wq1, bq1, qh, PD, PD, 0, BB*SQ*PD);
    gemm_f32_wmma<<<dim3(PD/256, NP/16, BB), 128, 0, stream>>>(
        memb, WkT, kmem, bk1, NP, PD, PD,
        (long)NP*PD, 0L, (long)NP*PD, 2, 0);
    gemm_f32_wmma<<<dim3(PD/256, NP/16, BB), 128, 0, stream>>>(
        memb, WvT, vmem, bv1, NP, PD, PD,
        (long)NP*PD, 0L, (long)NP*PD, 2, 0);
    attn_cross_kernel<<<dim3(SQ, NH, BB), 32, 0, stream>>>(qh, kmem, vmem, attn);
    proj_res_ln_kernel<<<dim3(SQ, BB), PD, 0, stream>>>(attn, Wo1, bo1, q,
                                                        ln2g + l*PD, ln2b + l*PD, q, PD);
    // feed-forward (tiny)
    rowproj_kernel<<<(BB*SQ*FFD)/256, 256, 0, stream>>>(q, Wff1 + (size_t)l*FFD*PD,
                                                        bff1 + l*FFD, ffbuf, PD, FFD, 1,
                                                        BB*SQ*FFD);
    proj_res_ln_kernel<<<dim3(SQ, BB), PD, 0, stream>>>(ffbuf, Wff2 + (size_t)l*PD*FFD,
                                                        bff2 + l*PD, q,
                                                        ln3g + l*PD, ln3b + l*PD, q, FFD);
  }

  final_kernel<<<dim3(SQ, BB), PD, 0, stream>>>(q, lnfg, lnfb, Wcm, bcm, (float*)d_out);
}